// QHNet_20839181320730
// MI455X (gfx1250) — compile-verified
//
#include <hip/hip_runtime.h>
#include <math.h>

typedef __attribute__((ext_vector_type(2))) float v2f;
typedef __attribute__((ext_vector_type(8))) float v8f;
typedef int v4i_ __attribute__((vector_size(4 * sizeof(int))));

// CDNA5 f32 WMMA: D(16x16,f32) = A(16x4,f32) * B(4x16,f32) + C
#define WMMA4(a, b, c) \
  __builtin_amdgcn_wmma_f32_16x16x4_f32(false, (a), false, (b), (short)0, (c), false, false)

#define LN2F 0.69314718055994531f

#define AS1 __attribute__((address_space(1)))
#define AS3 __attribute__((address_space(3)))

#if __has_builtin(__builtin_amdgcn_global_load_async_to_lds_b128) && \
    __has_builtin(__builtin_amdgcn_s_wait_asynccnt)
#define ASYNC_LDS 1
#endif

__device__ __forceinline__ void copy16_to_lds(const float* gsrc, float* ldst) {
#ifdef ASYNC_LDS
  __builtin_amdgcn_global_load_async_to_lds_b128((AS1 v4i_*)gsrc, (AS3 v4i_*)ldst, 0, 0);
#else
  *(float4*)ldst = *(const float4*)gsrc;
#endif
}
__device__ __forceinline__ void copy_wait() {
#ifdef ASYNC_LDS
  __builtin_amdgcn_s_wait_asynccnt(0);
#endif
}

// softplus(x) - ln2 via native v_exp_f32 / v_log_f32 (avoids precise log1pf blowup)
__device__ __forceinline__ float ssp_f(float x) {
  float sp = __logf(1.f + __expf(x));
  sp = (x > 20.f) ? x : sp;  // overflow guard (exp(20) ~ 4.8e8, still exact enough)
  return sp - LN2F;
}
__device__ __forceinline__ float silu_f(float x) {
  return __fdividef(x, 1.f + __expf(-x));
}

// ---------------------------------------------------------------------------
// Strided WMMA GEMM: C[m*ldc + n*ccs] = act(sum_k A[m*lda+k*acs]*B[k*ldb+n]*scale + bias[n])
// Each wave computes a 16 x (16*NT) tile; A fragment reused across NT WMMAs.
// All offsets 32-bit so loads use SADDR + voffset addressing.
// ---------------------------------------------------------------------------
template <int NT>
__global__ __launch_bounds__(256) void gemm_wmma_f32(
    const float* __restrict__ A, int lda, int acs,
    const float* __restrict__ B, int ldb,
    float* __restrict__ C, int ldc, int ccs,
    const float* __restrict__ bias,
    int M, int N, int K, float scale, int act) {
  const int lane = threadIdx.x & 31;
  const int wave = threadIdx.x >> 5;
  const int m0 = blockIdx.x * 16;
  const int n0 = (blockIdx.y * 8 + wave) * (16 * NT);
  if (m0 >= M || n0 >= N) return;

  const int ln16 = lane & 15;
  const int kg = (lane >> 4) << 1;  // 0 / 2

  int aoff = (m0 + ln16) * lda + kg * acs;  // A[m0+ln16][kg]
  int boff = kg * ldb + n0 + ln16;          // B[kg][n0+ln16]
  const int astep = 4 * acs;
  const int bstep = 4 * ldb;

  v8f acc[NT];
#pragma unroll
  for (int t = 0; t < NT; ++t) acc[t] = (v8f){};

#pragma unroll 2
  for (int k0 = 0; k0 < K; k0 += 4) {
    v2f a;
    a.x = A[aoff];
    a.y = A[aoff + acs];
#pragma unroll
    for (int t = 0; t < NT; ++t) {
      v2f b;
      b.x = B[boff + t * 16];
      b.y = B[boff + ldb + t * 16];
      acc[t] = WMMA4(a, b, acc[t]);
    }
    aoff += astep;
    boff += bstep;
  }

  const int rbase = (lane >> 4) * 8;
#pragma unroll
  for (int t = 0; t < NT; ++t) {
    const int col = n0 + t * 16 + ln16;
    const float bc = bias ? bias[col] : 0.f;
#pragma unroll
    for (int r = 0; r < 8; ++r) {
      float v = acc[t][r] * scale + bc;
      if (act == 1) v = silu_f(v);
      C[(m0 + rbase + r) * ldc + col * ccs] = v;
    }
  }
}

// ---------------------------------------------------------------------------
// f0 = [xs, vnorm]  (N x 192)
// ---------------------------------------------------------------------------
__global__ __launch_bounds__(256) void f0_kernel(const float* __restrict__ x,
                                                 float* __restrict__ f0, int n_nodes) {
  int idx = blockIdx.x * blockDim.x + threadIdx.x;
  if (idx >= n_nodes * 192) return;
  int n = idx / 192, k = idx - n * 192;
  if (k < 128) {
    f0[idx] = x[n * 320 + k];
  } else {
    const float* p = x + n * 320 + 128 + (k - 128) * 3;
    f0[idx] = sqrtf(p[0] * p[0] + p[1] * p[1] + p[2] * p[2] + 1e-12f);
  }
}

// xv2[n,u,c] = xv[n,u,c] * g[n,128+u]   (N x 64 x 3 contiguous)
__global__ __launch_bounds__(256) void xv2_kernel(const float* __restrict__ x,
                                                  const float* __restrict__ g,
                                                  float* __restrict__ xv2, int n_nodes) {
  int idx = blockIdx.x * blockDim.x + threadIdx.x;
  if (idx >= n_nodes * 192) return;
  int n = idx / 192, r = idx - n * 192, u = r / 3;
  xv2[idx] = x[n * 320 + 128 + r] * g[n * 192 + 128 + u];
}

// ---------------------------------------------------------------------------
// Fused edge kernel: 4 waves/block, 16 edges/wave.
//   1. async-stage s_dst (128) + edge_attr tile (16x32) into LDS; compute ip1 (64)
//   2. tf = ssp(edge_attr @ Wf1 / sqrt32)   via WMMA -> LDS (C->A re-fragment)
//   3. tl = ssp(s0 @ Wl1 / sqrt320)         via WMMA -> LDS
//   4. per 16-col tile of 384: wf,wl via WMMA, w = wf*wl/32,
//      uniform-branch scatter-add into ns (N,192) / nv (N,192,3)
// ---------------------------------------------------------------------------
#define EW 16
#define WPB 4

__global__ __launch_bounds__(128) void edge_kernel(
    const long long* __restrict__ edge_index, const float* __restrict__ edge_attr,
    const float* __restrict__ edge_sh, const float* __restrict__ pre_s,
    const float* __restrict__ pre_v, const float* __restrict__ xs3,
    const float* __restrict__ xv3, const float* __restrict__ Wf1,
    const float* __restrict__ Wf2, const float* __restrict__ Wl1,
    const float* __restrict__ Wl2, float* __restrict__ ns, float* __restrict__ nv,
    int E_) {
  __shared__ __align__(16) float s0_lds[WPB][EW][192];
  __shared__ __align__(16) float ea_lds[WPB][EW][32];
  __shared__ __align__(16) float tf_lds[WPB][EW][32];
  __shared__ __align__(16) float tl_lds[WPB][EW][32];

  const int lane = threadIdx.x & 31;
  const int wave = threadIdx.x >> 5;
  const int e0 = (blockIdx.x * WPB + wave) * EW;
  if (e0 >= E_) return;

  const int ln16 = lane & 15;
  const int kg = (lane >> 4) << 1;
  const int rbase = (lane >> 4) * 8;

  const float inv_sqrt32 = 0.17677669529663687f;
  const float inv_sqrt320 = 0.05590169943749474f;
  const float inv_sqrt3 = 0.57735026918962576f;
  const float c0 = 0.5f, c1 = 0.86602540378443865f;

  // Per-edge metadata in lanes 0..15
  int dst_i = 0, src_i = 0;
  float sh0_i = 0.f, s1x_i = 0.f, s1y_i = 0.f, s1z_i = 0.f;
  if (lane < EW) {
    dst_i = (int)edge_index[e0 + lane];
    src_i = (int)edge_index[E_ + e0 + lane];
    const float* sh = edge_sh + (e0 + lane) * 4;
    sh0_i = sh[0]; s1x_i = sh[1]; s1y_i = sh[2]; s1z_i = sh[3];
  }

  // ---- async-stage edge_attr tile (contiguous 512 floats) ----
#pragma unroll
  for (int c = 0; c < 4; ++c) {
    int q = c * 32 + lane;  // float4 chunk id
    copy16_to_lds(edge_attr + e0 * 32 + q * 4, &ea_lds[wave][0][0] + q * 4);
  }

  // ---- stage s0 tile: pre_s[dst] (async b128) + ip1 (computed) ----
  for (int i = 0; i < EW; ++i) {
    int d = __shfl(dst_i, i, 32);
    int s = __shfl(src_i, i, 32);
    copy16_to_lds(pre_s + d * 128 + lane * 4, &s0_lds[wave][i][lane * 4]);
    const float* pvd = pre_v + d * 192;
    const float* pvs = pre_v + s * 192;
#pragma unroll
    for (int u = lane; u < 64; u += 32) {
      float acc = pvd[u * 3] * pvs[u * 3] + pvd[u * 3 + 1] * pvs[u * 3 + 1] +
                  pvd[u * 3 + 2] * pvs[u * 3 + 2];
      s0_lds[wave][i][128 + u] = acc * (1.f / 3.f);
    }
  }
  copy_wait();

  // ---- tf = ssp(edge_attr @ Wf1 * inv_sqrt32) : 16x32 ----
#pragma unroll
  for (int nt = 0; nt < 2; ++nt) {
    v8f acc = {};
#pragma unroll
    for (int k0 = 0; k0 < 32; k0 += 4) {
      int k = k0 + kg;
      v2f a, b;
      a.x = ea_lds[wave][ln16][k];
      a.y = ea_lds[wave][ln16][k + 1];
      b.x = Wf1[k * 32 + nt * 16 + ln16];
      b.y = Wf1[(k + 1) * 32 + nt * 16 + ln16];
      acc = WMMA4(a, b, acc);
    }
#pragma unroll
    for (int r = 0; r < 8; ++r)
      tf_lds[wave][rbase + r][nt * 16 + ln16] = ssp_f(acc[r] * inv_sqrt32);
  }

  // ---- tl = ssp(s0 @ Wl1 * inv_sqrt320) : K=320 logical / 192 physical ----
#pragma unroll
  for (int nt = 0; nt < 2; ++nt) {
    v8f acc = {};
#pragma unroll 2
    for (int k0 = 0; k0 < 320; k0 += 4) {
      int k = k0 + kg;
      int kk = (k < 128) ? k : k - 128;  // [128,256)->s_dst again, [256,320)->ip1
      v2f a, b;
      a.x = s0_lds[wave][ln16][kk];
      a.y = s0_lds[wave][ln16][kk + 1];
      b.x = Wl1[k * 32 + nt * 16 + ln16];
      b.y = Wl1[(k + 1) * 32 + nt * 16 + ln16];
      acc = WMMA4(a, b, acc);
    }
#pragma unroll
    for (int r = 0; r < 8; ++r)
      tl_lds[wave][rbase + r][nt * 16 + ln16] = ssp_f(acc[r] * inv_sqrt320);
  }

  // ---- per 16-col tile: w = (tf@Wf2/sqrt32)*(tl@Wl2/sqrt32) -> scatter ----
  for (int nt = 0; nt < 24; ++nt) {
    v8f wf = {}, wl = {};
#pragma unroll
    for (int k0 = 0; k0 < 32; k0 += 4) {
      int k = k0 + kg;
      v2f a, b;
      a.x = tf_lds[wave][ln16][k];
      a.y = tf_lds[wave][ln16][k + 1];
      b.x = Wf2[k * 384 + nt * 16 + ln16];
      b.y = Wf2[(k + 1) * 384 + nt * 16 + ln16];
      wf = WMMA4(a, b, wf);
      a.x = tl_lds[wave][ln16][k];
      a.y = tl_lds[wave][ln16][k + 1];
      b.x = Wl2[k * 384 + nt * 16 + ln16];
      b.y = Wl2[(k + 1) * 384 + nt * 16 + ln16];
      wl = WMMA4(a, b, wl);
    }
    const int ch = nt * 16 + ln16;  // segment bounds are multiples of 16 -> uniform branch
#pragma unroll
    for (int r = 0; r < 8; ++r) {
      const int erow = rbase + r;
      const float w = (wf[r] * inv_sqrt32) * (wl[r] * inv_sqrt32);
      const int d = __shfl(dst_i, erow, 32);
      const int s = __shfl(src_i, erow, 32);
      const float sh0 = __shfl(sh0_i, erow, 32);
      const float s1x = __shfl(s1x_i, erow, 32);
      const float s1y = __shfl(s1y_i, erow, 32);
      const float s1z = __shfl(s1z_i, erow, 32);
      if (ch < 128) {  // out0a -> ns[:, ch]
        float hs = xs3[s * 128 + ch];
        atomicAdd(&ns[d * 192 + ch], c0 * w * hs * sh0);
      } else if (ch < 256) {  // out1a -> nv[:, j, :]
        int j = ch - 128;
        float hs = xs3[s * 128 + j];
        float base = c1 * w * hs;
        float* p = &nv[d * 576 + j * 3];
        atomicAdd(p + 0, base * s1x);
        atomicAdd(p + 1, base * s1y);
        atomicAdd(p + 2, base * s1z);
      } else if (ch < 320) {  // out1b -> nv[:, 128+j, :]
        int j = ch - 256;
        const float* hv = xv3 + s * 192 + j * 3;
        float base = c1 * w * sh0;
        float* p = &nv[d * 576 + (128 + j) * 3];
        atomicAdd(p + 0, base * hv[0]);
        atomicAdd(p + 1, base * hv[1]);
        atomicAdd(p + 2, base * hv[2]);
      } else {  // out0b -> ns[:, 128+j]
        int j = ch - 320;
        const float* hv = xv3 + s * 192 + j * 3;
        float dotv = hv[0] * s1x + hv[1] * s1y + hv[2] * s1z;
        atomicAdd(&ns[d * 192 + 128 + j], c0 * w * dotv * inv_sqrt3);
      }
    }
  }
}

// ---------------------------------------------------------------------------
static void launch_gemm(const float* A, int lda, int acs, const float* B, int ldb,
                        float* C, int ldc, int ccs, const float* bias, int M, int N,
                        int K, float scale, int act, hipStream_t stream) {
  int nt = (N % 64 == 0 && N >= 256) ? 4 : (N % 32 == 0 && N >= 128) ? 2 : 1;
  int tiles = N / (16 * nt);
  dim3 grid(M / 16, (tiles + 7) / 8);
  if (nt == 4)
    gemm_wmma_f32<4><<<grid, 256, 0, stream>>>(A, lda, acs, B, ldb, C, ldc, ccs, bias, M, N, K, scale, act);
  else if (nt == 2)
    gemm_wmma_f32<2><<<grid, 256, 0, stream>>>(A, lda, acs, B, ldb, C, ldc, ccs, bias, M, N, K, scale, act);
  else
    gemm_wmma_f32<1><<<grid, 256, 0, stream>>>(A, lda, acs, B, ldb, C, ldc, ccs, bias, M, N, K, scale, act);
}

extern "C" void kernel_launch(void* const* d_in, const int* in_sizes, int n_in,
                              void* d_out, int out_size, void* d_ws, size_t ws_size,
                              hipStream_t stream) {
  const float* x = (const float*)d_in[0];
  const long long* edge_index = (const long long*)d_in[1];
  const float* edge_attr = (const float*)d_in[2];
  const float* edge_sh = (const float*)d_in[3];
  const float* Wpre_s = (const float*)d_in[4];
  const float* bpre_s = (const float*)d_in[5];
  const float* Wpre_v = (const float*)d_in[6];
  const float* Wg1 = (const float*)d_in[7];
  const float* bg1 = (const float*)d_in[8];
  const float* Wg2 = (const float*)d_in[9];
  const float* bg2 = (const float*)d_in[10];
  const float* Wn_s = (const float*)d_in[11];
  const float* bn_s = (const float*)d_in[12];
  const float* Wn_v = (const float*)d_in[13];
  const float* Wf1 = (const float*)d_in[14];
  const float* Wf2 = (const float*)d_in[15];
  const float* Wl1 = (const float*)d_in[16];
  const float* Wl2 = (const float*)d_in[17];
  const float* Wo_s = (const float*)d_in[18];
  const float* bo_s = (const float*)d_in[19];
  const float* Wo_v = (const float*)d_in[20];
  float* out = (float*)d_out;

  const int N = in_sizes[0] / 320;
  const int E = in_sizes[1] / 2;

  const float is128 = 0.08838834764831845f;
  const float is64 = 0.125f;
  const float is192 = 0.07216878364870323f;

  float* ws = (float*)d_ws;
  size_t off = 0;
  float* pre_s = ws + off; off += (size_t)N * 128;
  float* pre_v = ws + off; off += (size_t)N * 192;
  float* f0    = ws + off; off += (size_t)N * 192;
  float* h1    = ws + off; off += (size_t)N * 192;
  float* g     = ws + off; off += (size_t)N * 192;
  float* xv2   = ws + off; off += (size_t)N * 192;
  float* xs3   = ws + off; off += (size_t)N * 128;
  float* xv3   = ws + off; off += (size_t)N * 192;
  float* ns    = ws + off; off += (size_t)N * 192;
  float* nv    = ws + off; off += (size_t)N * 576;

  (void)hipMemsetAsync(ns, 0, (size_t)N * 768 * sizeof(float), stream);

  // pre_s = xs @ Wpre_s / sqrt(128) + bpre_s
  launch_gemm(x, 320, 1, Wpre_s, 128, pre_s, 128, 1, bpre_s, N, 128, 128, is128, 0, stream);
  // pre_v[:,:,c] = xv[:,:,c] @ Wpre_v / sqrt(64)
  for (int c = 0; c < 3; ++c)
    launch_gemm(x + 128 + c, 320, 3, Wpre_v, 64, pre_v + c, 192, 3, nullptr, N, 64, 64,
                is64, 0, stream);

  {
    int total = N * 192;
    f0_kernel<<<(total + 255) / 256, 256, 0, stream>>>(x, f0, N);
  }
  launch_gemm(f0, 192, 1, Wg1, 192, h1, 192, 1, bg1, N, 192, 192, 1.f, 1, stream);
  launch_gemm(h1, 192, 1, Wg2, 192, g, 192, 1, bg2, N, 192, 192, 1.f, 0, stream);

  launch_gemm(g, 192, 1, Wn_s, 128, xs3, 128, 1, bn_s, N, 128, 128, is128, 0, stream);
  {
    int total = N * 192;
    xv2_kernel<<<(total + 255) / 256, 256, 0, stream>>>(x, g, xv2, N);
  }
  for (int c = 0; c < 3; ++c)
    launch_gemm(xv2 + c, 192, 3, Wn_v, 64, xv3 + c, 192, 3, nullptr, N, 64, 64, is64, 0,
                stream);

  {
    int blocks = (E + WPB * EW - 1) / (WPB * EW);
    edge_kernel<<<blocks, WPB * 32, 0, stream>>>(edge_index, edge_attr, edge_sh, pre_s,
                                                 pre_v, xs3, xv3, Wf1, Wf2, Wl1, Wl2, ns,
                                                 nv, E);
  }

  launch_gemm(ns, 192, 1, Wo_s, 256, out, 640, 1, bo_s, N, 256, 192, is192, 0, stream);
  for (int c = 0; c < 3; ++c)
    launch_gemm(nv + c, 576, 3, Wo_v, 128, out + 256 + c, 640, 3, nullptr, N, 128, 192,
                is192, 0, stream);
}